// WESUP_54992761258930
// MI455X (gfx1250) — compile-verified
//
#include <hip/hip_runtime.h>
#include <math.h>

#define C_FEAT 4224
#define NPIX   65536   // 256*256
#define NSEG   160
#define CC     48      // channels per segment-reduce block (4224 = 48*88)
#define PP     8192    // pixels per segment-reduce block

typedef __attribute__((ext_vector_type(2))) float v2f;
typedef __attribute__((ext_vector_type(8))) float v8f;

// ---------------- zero init ----------------
__global__ void zero_kernel(float* p, int n) {
    int i = blockIdx.x * blockDim.x + threadIdx.x;
    if (i < n) p[i] = 0.0f;
}

// ---------------- per-segment pixel & class counts (exact int atomics) ----------------
__global__ void count_kernel(const int* __restrict__ sp, const int* __restrict__ y,
                             int* __restrict__ cnt, int* __restrict__ ccls) {
    int i = blockIdx.x * blockDim.x + threadIdx.x;
    if (i < NPIX) {
        int s = sp[i];
        atomicAdd(&cnt[s], 1);
        atomicAdd(&ccls[s * 3 + y[i]], 1);
    }
}

// ---------------- segment feature sums: LDS binning + global atomic flush ----------------
// Streams feat_map (1.08 GB) exactly once. Each block: 48 channels x 8192 pixels.
// Segment IDs staged in LDS once, reused for 4 channels per inner iteration
// (1 ds_load_b128 per 16 ds_add_f32). acc layout acc[cl*160+s] keeps concurrent
// ds_add bank index ~ segment id (mod 64) -> low conflict. global_prefetch_b8
// hints cover the next 4 channel rows (256 thr x 128B = 32KB per row chunk).
__global__ void seg_reduce_kernel(const float* __restrict__ feat_map,
                                  const int* __restrict__ sp,
                                  float* __restrict__ sums) {
    __shared__ float acc[CC * NSEG];   // 30 KB
    __shared__ int   sseg[PP];         // 32 KB
    const int tid = threadIdx.x;
    const int c0 = blockIdx.x * CC;
    const int p0 = blockIdx.y * PP;

    for (int i = tid; i < CC * NSEG; i += 256) acc[i] = 0.0f;
    for (int i = tid; i < PP / 4; i += 256)
        ((int4*)sseg)[i] = ((const int4*)(sp + p0))[i];
    __syncthreads();

    for (int cl = 0; cl < CC; cl += 4) {
        const float* __restrict__ r0 = feat_map + (size_t)(c0 + cl) * NPIX + p0;
        const float* __restrict__ r1 = r0 + NPIX;
        const float* __restrict__ r2 = r0 + 2 * (size_t)NPIX;
        const float* __restrict__ r3 = r0 + 3 * (size_t)NPIX;
        float* __restrict__ a0 = acc + (cl + 0) * NSEG;
        float* __restrict__ a1 = acc + (cl + 1) * NSEG;
        float* __restrict__ a2 = acc + (cl + 2) * NSEG;
        float* __restrict__ a3 = acc + (cl + 3) * NSEG;

        // Prefetch next 4 channel rows for this pixel chunk (speculative, safe).
        __builtin_prefetch(r0 + 4 * (size_t)NPIX + tid * 32, 0, 0);
        __builtin_prefetch(r1 + 4 * (size_t)NPIX + tid * 32, 0, 0);
        __builtin_prefetch(r2 + 4 * (size_t)NPIX + tid * 32, 0, 0);
        __builtin_prefetch(r3 + 4 * (size_t)NPIX + tid * 32, 0, 0);

        for (int p = tid * 4; p < PP; p += 256 * 4) {
            float4 v0 = ((const float4*)r0)[p >> 2];
            float4 v1 = ((const float4*)r1)[p >> 2];
            float4 v2 = ((const float4*)r2)[p >> 2];
            float4 v3 = ((const float4*)r3)[p >> 2];
            int4   s  = ((const int4*)sseg)[p >> 2];
            atomicAdd(&a0[s.x], v0.x);
            atomicAdd(&a0[s.y], v0.y);
            atomicAdd(&a0[s.z], v0.z);
            atomicAdd(&a0[s.w], v0.w);
            atomicAdd(&a1[s.x], v1.x);
            atomicAdd(&a1[s.y], v1.y);
            atomicAdd(&a1[s.z], v1.z);
            atomicAdd(&a1[s.w], v1.w);
            atomicAdd(&a2[s.x], v2.x);
            atomicAdd(&a2[s.y], v2.y);
            atomicAdd(&a2[s.z], v2.z);
            atomicAdd(&a2[s.w], v2.w);
            atomicAdd(&a3[s.x], v3.x);
            atomicAdd(&a3[s.y], v3.y);
            atomicAdd(&a3[s.z], v3.z);
            atomicAdd(&a3[s.w], v3.w);
        }
    }
    __syncthreads();

    for (int i = tid; i < CC * NSEG; i += 256) {
        int cl = i / NSEG, s = i % NSEG;
        atomicAdd(&sums[(size_t)s * C_FEAT + c0 + cl], acc[i]);
    }
}

// ---------------- mean feature + L2-normalized feature ----------------
__global__ void featnorm_kernel(const float* __restrict__ sums, const int* __restrict__ cnt,
                                float* __restrict__ feat, float* __restrict__ nf) {
    int s = blockIdx.x;
    int tid = threadIdx.x;
    __shared__ float red[256];
    float inv = 1.0f / fmaxf((float)cnt[s], 1.0f);
    float ss = 0.0f;
    for (int c = tid; c < C_FEAT; c += 256) {
        float f = sums[(size_t)s * C_FEAT + c] * inv;
        feat[(size_t)s * C_FEAT + c] = f;
        ss += f * f;
    }
    red[tid] = ss;
    __syncthreads();
    for (int o = 128; o > 0; o >>= 1) {
        if (tid < o) red[tid] += red[tid + o];
        __syncthreads();
    }
    float inv2 = 1.0f / fmaxf(sqrtf(red[0]), 1e-12f);
    for (int c = tid; c < C_FEAT; c += 256)
        nf[(size_t)s * C_FEAT + c] = feat[(size_t)s * C_FEAT + c] * inv2;
}

// ---------------- initial labels (argmax with first-max ties, majority rules) ----------------
__global__ void label_init_kernel(const int* __restrict__ cnt, const int* __restrict__ ccls,
                                  int* __restrict__ labels0) {
    int s = threadIdx.x;
    if (s < NSEG) {
        int n0 = ccls[s * 3 + 0], n1 = ccls[s * 3 + 1], n2 = ccls[s * 3 + 2];
        int lab = 0, best = n0;
        if (n1 > best) { best = n1; lab = 1; }
        if (n2 > best) { best = n2; lab = 2; }
        float half = floorf((float)cnt[s] * 0.5f);
        int lab0 = ((float)n2 > half) ? 2 : ((n1 >= 1) ? 1 : 0);
        labels0[s] = (lab == 0) ? lab0 : lab;
    }
}

// ---------------- cosine-similarity label propagation ----------------
__global__ void sim_kernel(const float* __restrict__ nf, const int* __restrict__ labels0,
                           int* __restrict__ labels1) {
    int i = blockIdx.x;
    int j = threadIdx.x;
    __shared__ float Arow[NSEG];
    if (j < NSEG) {
        const float4* __restrict__ ri = (const float4*)(nf + (size_t)i * C_FEAT);
        const float4* __restrict__ rj = (const float4*)(nf + (size_t)j * C_FEAT);
        float a = 0.0f;
        for (int k = 0; k < C_FEAT / 4; ++k) {
            float4 x = ri[k];
            float4 yv = rj[k];
            a += x.x * yv.x + x.y * yv.y + x.z * yv.z + x.w * yv.w;
        }
        Arow[j] = (labels0[j] != 0) ? a : -2.0f;
    }
    __syncthreads();
    if (j == 0) {
        float maxv = Arow[0];
        int midx = 0;
        for (int t = 1; t < NSEG; ++t)
            if (Arow[t] > maxv) { maxv = Arow[t]; midx = t; }
        int li = labels0[i];
        labels1[i] = (li == 0 && maxv >= 0.8f) ? labels0[midx] : li;
    }
}

// ---------------- f32 WMMA GEMM with fused bias + ReLU ----------------
// C[M,N] = relu(A[M,K] @ B[K,N] + bias[N]); M,N mult of 16, K mult of 4.
// Block = 128 threads (4 waves), each wave owns one 16x16 tile; block covers 64 cols.
__global__ void gemm_relu_wmma(const float* __restrict__ A, const float* __restrict__ B,
                               const float* __restrict__ bias, float* __restrict__ Cout,
                               int N, int K, int do_relu) {
    int wave  = threadIdx.x >> 5;
    int lane  = threadIdx.x & 31;
    int n0    = blockIdx.x * 64 + wave * 16;
    int m0    = blockIdx.y * 16;
    if (n0 >= N) return;
    int lhalf = lane >> 4;     // 0: K pair {0,1}, 1: K pair {2,3}
    int l16   = lane & 15;

    const float* __restrict__ arow = A + (size_t)(m0 + l16) * K + 2 * lhalf;
    const float* __restrict__ bcol = B + (size_t)(2 * lhalf) * N + n0 + l16;

    v8f acc = {};
    for (int k0 = 0; k0 < K; k0 += 4) {
        v2f a, b;
        a.x = arow[k0];
        a.y = arow[k0 + 1];
        b.x = bcol[(size_t)k0 * N];
        b.y = bcol[(size_t)(k0 + 1) * N];
        acc = __builtin_amdgcn_wmma_f32_16x16x4_f32(
            /*neg_a=*/false, a, /*neg_b=*/false, b,
            /*c_mod=*/(short)0, acc, /*reuse_a=*/false, /*reuse_b=*/false);
    }
    float bv = bias[n0 + l16];
    for (int v = 0; v < 8; ++v) {
        float val = acc[v] + bv;
        if (do_relu) val = fmaxf(val, 0.0f);
        Cout[(size_t)(m0 + v + 8 * lhalf) * N + n0 + l16] = val;
    }
}

// ---------------- classifier head: (160,32)@(32,2)+bc -> softmax; emit pred + labels ----------------
__global__ void head_kernel(const float* __restrict__ h3, const float* __restrict__ wc,
                            const float* __restrict__ bc, const int* __restrict__ labels1,
                            float* __restrict__ out) {
    int i = threadIdx.x;
    if (i < NSEG) {
        float z0 = bc[0], z1 = bc[1];
        for (int k = 0; k < 32; ++k) {
            float h = h3[i * 32 + k];
            z0 += h * wc[k * 2 + 0];
            z1 += h * wc[k * 2 + 1];
        }
        float m  = fmaxf(z0, z1);
        float e0 = expf(z0 - m), e1 = expf(z1 - m);
        float inv = 1.0f / (e0 + e1);
        out[2 * i]     = e0 * inv;
        out[2 * i + 1] = e1 * inv;
        out[2 * NSEG + i] = (float)labels1[i];
    }
}

extern "C" void kernel_launch(void* const* d_in, const int* in_sizes, int n_in,
                              void* d_out, int out_size, void* d_ws, size_t ws_size,
                              hipStream_t stream) {
    const float* feat_map = (const float*)d_in[0];
    const int*   sp       = (const int*)d_in[1];
    const int*   y        = (const int*)d_in[2];
    const float* w1       = (const float*)d_in[3];
    const float* b1       = (const float*)d_in[4];
    const float* w2       = (const float*)d_in[5];
    const float* b2       = (const float*)d_in[6];
    const float* w3       = (const float*)d_in[7];
    const float* b3       = (const float*)d_in[8];
    const float* wc       = (const float*)d_in[9];
    const float* bc       = (const float*)d_in[10];
    float* out = (float*)d_out;

    char* base = (char*)d_ws;
    size_t off = 0;
    auto take = [&](size_t bytes) -> void* {
        off = (off + 255) & ~(size_t)255;
        void* p = base + off;
        off += bytes;
        return p;
    };
    float* sums    = (float*)take((size_t)NSEG * C_FEAT * 4);
    int*   cnt     = (int*)  take((size_t)NSEG * 4);
    int*   ccls    = (int*)  take((size_t)NSEG * 3 * 4);
    int*   labels0 = (int*)  take((size_t)NSEG * 4);
    int*   labels1 = (int*)  take((size_t)NSEG * 4);
    float* feat    = (float*)take((size_t)NSEG * C_FEAT * 4);
    float* nf      = (float*)take((size_t)NSEG * C_FEAT * 4);
    float* h1      = (float*)take((size_t)NSEG * 1024 * 4);
    float* h2      = (float*)take((size_t)NSEG * 1024 * 4);
    float* h3      = (float*)take((size_t)NSEG * 32 * 4);

    // zero accumulators (deterministic re-init each call)
    {
        int n = NSEG * C_FEAT;
        zero_kernel<<<(n + 255) / 256, 256, 0, stream>>>(sums, n);
        zero_kernel<<<1, 256, 0, stream>>>((float*)cnt, NSEG);
        zero_kernel<<<2, 256, 0, stream>>>((float*)ccls, NSEG * 3);
    }

    // counts
    count_kernel<<<NPIX / 256, 256, 0, stream>>>(sp, y, cnt, ccls);

    // segment feature sums (streams the 1.08 GB feat_map once)
    {
        dim3 grid(C_FEAT / CC, NPIX / PP);  // (88, 8)
        seg_reduce_kernel<<<grid, 256, 0, stream>>>(feat_map, sp, sums);
    }

    // means + normalized features
    featnorm_kernel<<<NSEG, 256, 0, stream>>>(sums, cnt, feat, nf);

    // initial labels
    label_init_kernel<<<1, 192, 0, stream>>>(cnt, ccls, labels0);

    // similarity-based propagation
    sim_kernel<<<NSEG, 192, 0, stream>>>(nf, labels0, labels1);

    // MLP via f32 WMMA
    gemm_relu_wmma<<<dim3(1024 / 64, NSEG / 16), 128, 0, stream>>>(feat, w1, b1, h1, 1024, C_FEAT, 1);
    gemm_relu_wmma<<<dim3(1024 / 64, NSEG / 16), 128, 0, stream>>>(h1, w2, b2, h2, 1024, 1024, 1);
    gemm_relu_wmma<<<dim3(1,         NSEG / 16), 128, 0, stream>>>(h2, w3, b3, h3, 32, 1024, 1);

    // softmax head + label output
    head_kernel<<<1, 192, 0, stream>>>(h3, wc, bc, labels1, out);
}